// EndtoEndModel_50534585205261
// MI455X (gfx1250) — compile-verified
//
#include <hip/hip_runtime.h>
#include <cmath>

typedef __attribute__((ext_vector_type(2))) float v2f;
typedef __attribute__((ext_vector_type(8))) float v8f;
typedef __attribute__((ext_vector_type(4))) int   v4i;

constexpr int B   = 256, T = 256, I = 16, H = 256, O = 11;
constexpr int HYP = 512, BLDD = 8;
constexpr int N1   = H * I + H * H + 2 * H;   // 70144
constexpr int N2   = O * H + O;               // 2827
constexpr int NTOT = N1 + N2;                 // 72971

// d_out layout (flat float32, reference return order)
constexpr long long OFF_ACT  = 0;                                  // (B,T,3)
constexpr long long OFF_BLDI = (long long)B * T * 3;               // (B,T,8)
constexpr long long OFF_H    = OFF_BLDI + (long long)B * T * 8;    // (1,B,H)
constexpr long long OFF_P    = OFF_H + (long long)B * H;           // (B,NTOT)

// ---------- CDNA5 async global->LDS staging (guarded, with fallback) ---------
#if defined(__has_builtin)
#if __has_builtin(__builtin_amdgcn_global_load_async_to_lds_b32) && \
    __has_builtin(__builtin_amdgcn_global_load_async_to_lds_b128)
#define HAVE_ASYNC_LDS 1
#endif
#endif

#ifdef HAVE_ASYNC_LDS
typedef __attribute__((address_space(1))) int  g_i32;
typedef __attribute__((address_space(3))) int  l_i32;
typedef __attribute__((address_space(1))) v4i  g_v4i;
typedef __attribute__((address_space(3))) v4i  l_v4i;
#endif

__device__ __forceinline__ void async_copy_b32(const float* gsrc, float* ldst) {
#ifdef HAVE_ASYNC_LDS
  // per-lane global addr -> per-lane LDS addr (supports padded scatter)
  __builtin_amdgcn_global_load_async_to_lds_b32(
      (g_i32*)(uintptr_t)gsrc,
      (l_i32*)(uint32_t)(uintptr_t)ldst, 0, 0);
#else
  *ldst = *gsrc;
#endif
}

__device__ __forceinline__ void async_copy_b128(const float* gsrc, float* ldst) {
#ifdef HAVE_ASYNC_LDS
  __builtin_amdgcn_global_load_async_to_lds_b128(
      (g_v4i*)(uintptr_t)gsrc,
      (l_v4i*)(uint32_t)(uintptr_t)ldst, 0, 0);
#else
  *(float4*)ldst = *(const float4*)gsrc;
#endif
}

__device__ __forceinline__ void async_staging_wait() {
#ifdef HAVE_ASYNC_LDS
  asm volatile("s_wait_asynccnt 0x0" ::: "memory");
#endif
}

// ---------------- Kernel 1a: hidden = relu(bld @ W1 + b1) --------------------
__global__ void hyper_hidden_kernel(const float* __restrict__ bld,
                                    const float* __restrict__ W1,
                                    const float* __restrict__ b1,
                                    float* __restrict__ hid) {
  int idx = blockIdx.x * blockDim.x + threadIdx.x;   // B*HYP threads
  int r = idx / HYP, c = idx % HYP;
  float s = b1[c];
#pragma unroll
  for (int j = 0; j < BLDD; ++j) s = fmaf(bld[r * BLDD + j], W1[j * HYP + c], s);
  hid[idx] = fmaxf(s, 0.0f);
}

// ------- Kernel 1b: params = hidden @ W2 + b2  (fp32 WMMA 16x16x4) -----------
__global__ void hyper_gemm_kernel(const float* __restrict__ Hm,
                                  const float* __restrict__ W2,
                                  const float* __restrict__ b2,
                                  float* __restrict__ Pout) {
  constexpr int TILES_N = (NTOT + 15) / 16;          // 4561
  const int lane = threadIdx.x & 31;
  const int wave = threadIdx.x >> 5;
  const int tile = blockIdx.x * 8 + wave;
  const int tm = tile / TILES_N;
  const int tn = tile % TILES_N;
  const int m0 = tm * 16;
  const int n0 = tn * 16;

  const int mrow  = lane & 15;            // A row this lane holds
  const int khalf = (lane >> 4) << 1;     // lanes 0-15: K pair {0,1}; 16-31: {2,3}
  const int col   = n0 + (lane & 15);     // B/C column this lane holds
  const int colc  = col < NTOT ? col : NTOT - 1;   // clamp loads, guard stores

  v8f c = {};
  const float* arow = Hm + (long long)(m0 + mrow) * HYP;
  for (int k = 0; k < HYP; k += 4) {
    v2f a, bm;
    a.x  = arow[k + khalf];
    a.y  = arow[k + khalf + 1];
    bm.x = W2[(long long)(k + khalf)     * NTOT + colc];
    bm.y = W2[(long long)(k + khalf + 1) * NTOT + colc];
    c = __builtin_amdgcn_wmma_f32_16x16x4_f32(false, a, false, bm,
                                              (short)0, c, false, false);
  }

  if (col < NTOT) {
    const float bb = b2[col];
#pragma unroll
    for (int r = 0; r < 8; ++r) {
      int m = m0 + r + ((lane >> 4) << 3);   // VGPR r: lanes0-15 M=r, 16-31 M=r+8
      Pout[(long long)m * NTOT + col] = c[r] + bb;
    }
  }
}

// ---------------- Kernel 2: fused per-batch RNN scan -------------------------
// One workgroup (256 thr) per batch element; 316.5 KB of LDS per WG (CDNA5-only).
constexpr int L_WHT  = 0;                    // W_hh^T, [h][g], stride 257 (pad)
constexpr int L_WIHT = L_WHT + 256 * 257;    // W_ih^T, [i][g]           (4096)
constexpr int L_WOUT = L_WIHT + 16 * 256;    // W_out padded [h][16]     (4096)
constexpr int L_STIM = L_WOUT + 256 * 16;    // stimulus [t][i]          (4096)
constexpr int L_BSUM = L_STIM + 256 * 16;    // b_ih+b_hh                 (256)
constexpr int L_BOUT = L_BSUM + 256;         // b_out padded               (16)
constexpr int L_HBUF = L_BOUT + 16;          // h double buffer           (512)
constexpr int L_PART = L_HBUF + 512;         // projection partials       (256)
constexpr int LDS_FLOATS = L_PART + 256;     // 79120 floats = 316480 B < 320 KB

__global__ void rnn_fused_kernel(const float* __restrict__ stimulus,
                                 const float* __restrict__ Pall,
                                 float* __restrict__ act_out,
                                 float* __restrict__ bldi_out,
                                 float* __restrict__ h_out) {
  extern __shared__ float lds[];
  const int b   = blockIdx.x;
  const int tid = threadIdx.x;
  const float* P = Pall + (long long)b * NTOT;

  // ---- W_hh transposed: read [g][h] coalesced, async-scatter to [h*257+g]
  for (int i = 0; i < H; ++i) {
    int src = H * I + i * H + tid;                  // W_hh[g=i][h=tid]
    if (src + 2048 < N1) __builtin_prefetch(&P[src + 2048], 0, 1);
    async_copy_b32(&P[src], &lds[L_WHT + tid * 257 + i]);
  }
  // ---- W_ih^T: [i][g] (async scatter)
  for (int idx = tid; idx < H * I; idx += 256)
    async_copy_b32(&P[idx], &lds[L_WIHT + (idx & 15) * 256 + (idx >> 4)]);
  // ---- stimulus slice (T,I): contiguous, async b128 (16B per lane per op)
  for (int idx4 = tid * 4; idx4 < T * I; idx4 += 256 * 4)
    async_copy_b128(&stimulus[(long long)b * T * I + idx4], &lds[L_STIM + idx4]);
  // ---- W_out padded [h][16] (single predicated pass, regular ds stores)
  for (int idx = tid; idx < H * 16; idx += 256) {
    int hh = idx >> 4, o = idx & 15;
    lds[L_WOUT + idx] = (o < O) ? P[N1 + o * H + hh] : 0.0f;
  }
  // ---- biases, h0
  lds[L_BSUM + tid] = P[H * I + H * H + tid] + P[H * I + H * H + H + tid];
  if (tid < 16) lds[L_BOUT + tid] = (tid < O) ? P[N1 + O * H + tid] : 0.0f;
  lds[L_HBUF + tid] = 0.0f;                        // buffer 0 = h_{-1} = 0
  async_staging_wait();                            // this wave's async copies done
  __syncthreads();                                 // all waves' staging visible

  const int g = tid;            // output unit for the matvec
  const int o = tid & 15;       // projection output slot
  const int j = tid >> 4;       // projection partial index

  for (int t = 0; t < T; ++t) {
    const float* hprev = lds + L_HBUF + ((t & 1) ? 256 : 0);
    float*       hcur  = lds + L_HBUF + ((t & 1) ? 0 : 256);

    // x_proj on the fly: rnn_input[t] = (t==0) ? 0 : stimulus[t-1]
    float acc = lds[L_BSUM + g];
    if (t > 0) {
      const float* xv = lds + L_STIM + (t - 1) * 16;
#pragma unroll
      for (int i = 0; i < 16; ++i) acc = fmaf(xv[i], lds[L_WIHT + i * 256 + g], acc);
    }
    // recurrent matvec from LDS (broadcast hprev, conflict-free W column reads)
#pragma unroll 8
    for (int hh = 0; hh < H; ++hh)
      acc = fmaf(hprev[hh], lds[L_WHT + hh * 257 + g], acc);

    float hn = tanhf(acc);
    hcur[g] = hn;
    __syncthreads();                               // hcur complete

    // output projection: 16x16 partial sums, then 16-lane reduce
    float p = 0.0f;
#pragma unroll
    for (int k = 0; k < 16; ++k)
      p = fmaf(hcur[j * 16 + k], lds[L_WOUT + (j * 16 + k) * 16 + o], p);
    lds[L_PART + o * 16 + j] = p;
    __syncthreads();                               // partials complete

    if (tid < O) {
      float s = lds[L_BOUT + tid];
#pragma unroll
      for (int jj = 0; jj < 16; ++jj) s += lds[L_PART + tid * 16 + jj];
      // nan_to_num
      if (isnan(s)) s = 0.0f;
      s = fminf(fmaxf(s, -3.402823466e38f), 3.402823466e38f);
      if (tid < 3)
        act_out[((long long)b * T + t) * 3 + tid] = 1.0f / (1.0f + expf(-s));
      else
        bldi_out[((long long)b * T + t) * 8 + (tid - 3)] = s;
    }
    __syncthreads();                               // protect partials / hprev swap
  }
  // final hidden state: t=255 wrote buffer 0
  h_out[(long long)b * H + g] = lds[L_HBUF + g];
}

// -----------------------------------------------------------------------------
extern "C" void kernel_launch(void* const* d_in, const int* in_sizes, int n_in,
                              void* d_out, int out_size, void* d_ws, size_t ws_size,
                              hipStream_t stream) {
  const float* bld  = (const float*)d_in[0];   // (B, 8)
  const float* stim = (const float*)d_in[1];   // (B, T, I)
  const float* W1   = (const float*)d_in[2];   // (8, 512)
  const float* b1   = (const float*)d_in[3];   // (512,)
  const float* W2   = (const float*)d_in[4];   // (512, NTOT)
  const float* b2   = (const float*)d_in[5];   // (NTOT,)
  float* out = (float*)d_out;
  float* hid = (float*)d_ws;                   // B*HYP floats = 512 KB scratch

  // 1a: hypernet hidden layer
  hyper_hidden_kernel<<<(B * HYP) / 256, 256, 0, stream>>>(bld, W1, b1, hid);

  // 1b: fp32 WMMA GEMM -> params written straight into d_out params region
  constexpr int TILES = 16 * ((NTOT + 15) / 16);   // 72976 tiles, 8 waves/block
  hyper_gemm_kernel<<<TILES / 8, 256, 0, stream>>>(hid, W2, b2, out + OFF_P);

  // 2: fused per-batch RNN scan out of 316.5 KB LDS
  rnn_fused_kernel<<<B, 256, LDS_FLOATS * sizeof(float), stream>>>(
      stim, out + OFF_P, out + OFF_ACT, out + OFF_BLDI, out + OFF_H);
}